// WaveletModel_28948079575227
// MI455X (gfx1250) — compile-verified
//
#include <hip/hip_runtime.h>

// CDNA5 / gfx1250 — wave32, WMMA f32 16x16x4 matrix pipe.
typedef __attribute__((ext_vector_type(2))) float v2f;
typedef __attribute__((ext_vector_type(8))) float v8f;

#define WAVES 8
#define BLOCK (WAVES * 32)

// ---------------------------------------------------------------------------
// Forward DWT level. Per wave: one 16x16x48 GEMM chain (12 x V_WMMA_F32_16X16X4_F32)
//   D[i,j] = sum_k A[i,k]*B[k,j]
//   A[i,k] = xp[2*t0 + 16*i + k]          (288-float circular window in LDS)
//   B[k,j] = h_{j&1}[k - (2*(j>>1)+1)]    (zero-padded LDS filter tables)
//   D[i,j] = y_{j&1}[t0 + 8*i + (j>>1)]   -> 128 approx + 128 detail outputs
// Filters (from ho, length 32):
//   analysis lowpass  ha[q] = ho[q]
//   analysis highpass hg[q] = signs[q]*ho[31-q], signs[q]=(-1)^(q+1)
// ---------------------------------------------------------------------------
__global__ __launch_bounds__(BLOCK) void dwt_fwd_level(
    const float* __restrict__ inB, int inStride, int inOff,
    float* __restrict__ aB, int aStride, int aOff,
    float* __restrict__ dB, int dStride, int dOff,
    const float* __restrict__ ho,
    int Nin, int ntiles)
{
  __shared__ float sFa[64];            // ha padded: sFa[16+q] = ho[q], else 0
  __shared__ float sFg[64];            // hg padded
  __shared__ float sSig[WAVES][304];

  const int tid  = threadIdx.x;
  const int wave = tid >> 5;
  const int lane = tid & 31;
  const int b    = blockIdx.y;
  const int tile = blockIdx.x * WAVES + wave;
  const bool active = tile < ntiles;
  const int t0 = tile * 128;

  if (tid < 64) {                      // build zero-padded filter tables
    const int q = tid - 16;
    float fa = 0.f, fg = 0.f;
    if (q >= 0 && q < 32) {
      fa = ho[q];
      const float hg = ho[31 - q];
      fg = (q & 1) ? hg : -hg;
    }
    sFa[tid] = fa;
    sFg[tid] = fg;
  }

  if (active) {
    const float* in = inB + (size_t)b * inStride + inOff;
    __builtin_prefetch(in + ((2 * t0) >= Nin ? 0 : 2 * t0), 0, 3);
    for (int s = lane; s < 288; s += 32) {      // 9 iters, wave-uniform
      int g = 2 * t0 + s - 16;                   // xp[m] = a[(m-16) mod Nin]
      if (g < 0)    g += Nin;
      if (g >= Nin) g -= Nin;
      sSig[wave][s] = in[g];
    }
  }
  __syncthreads();
  if (!active) return;

  const float* S = sSig[wave];
  const int row  = lane & 15;          // M-row (A) / N-col (B,D) owned by lane
  const int col  = row;
  const int hi   = lane >> 4;
  const int koff = hi << 1;            // K sub-lane split per ISA A/B layout
  const int f    = col & 1;            // 0 -> lowpass (approx), 1 -> highpass (detail)
  const int sft  = ((col >> 1) << 1) + 1;     // odd shift 2p+1 in [1,15]

  const float* T   = f ? sFg : sFa;    // this lane's (padded) filter column
  const int   aIx  = 16 * row + koff;  // A-pair base in staged window
  const int   bIx  = 16 + koff - sft;  // B-pair base in padded table (>= 1)

  v8f acc = {};
#if __has_builtin(__builtin_amdgcn_wmma_f32_16x16x4_f32)
  #pragma unroll
  for (int c = 0; c < 12; ++c) {
    v2f av = { S[aIx + 4 * c], S[aIx + 4 * c + 1] };
    v2f bv = { T[bIx + 4 * c], T[bIx + 4 * c + 1] };
    acc = __builtin_amdgcn_wmma_f32_16x16x4_f32(false, av, false, bv,
                                                (short)0, acc, false, false);
  }
#else
  const int p = col >> 1;
  #pragma unroll
  for (int r = 0; r < 8; ++r) {
    const int M = r + 8 * hi;
    const int base = 16 * M + 2 * p + 1;
    float sum = 0.f;
    for (int k = 0; k < 32; ++k) sum += S[base + k] * T[16 + k];
    acc[r] = sum;
  }
#endif

  float* aP = aB + (size_t)b * aStride + aOff;
  float* dP = dB + (size_t)b * dStride + dOff;
  float* P  = f ? dP : aP;             // branch-free stream select
  #pragma unroll
  for (int r = 0; r < 8; ++r) {
    const int t = t0 + 8 * (r + 8 * hi) + (col >> 1);
    P[t] = acc[r];
  }
}

// ---------------------------------------------------------------------------
// Inverse DWT level. Per wave: one 16x16x96 GEMM chain (24 WMMAs) via
// K-concatenation of the two upsampled streams:
//   out[m0 + 16*i + p] = sum_k ua[..]*Ho[k] + ud[..]*Go[k]
//   A[i,k] = up_{a|d}[m0 + 16*i + k] (circular, zeros at odd phase)
//   B[k,p] = Ho[k-p]  (first 48)  /  Go[k-p]  (second 48), padded LDS tables
// Synthesis filters: Ho[q] = ho[31-q];  Go[q] = (-1)^q * ho[q]
// ---------------------------------------------------------------------------
__global__ __launch_bounds__(BLOCK) void idwt_level(
    const float* __restrict__ apB, int apStride, int apOff,
    const float* __restrict__ dsB, int dsStride, int dsOff,
    float* __restrict__ outB, int outStride, int outOff,
    const float* __restrict__ ho,
    int two_n, int ntiles)
{
  __shared__ float sFH[64];            // Ho padded
  __shared__ float sFG[64];            // Go padded
  __shared__ float sA[WAVES][304];
  __shared__ float sD[WAVES][304];

  const int tid  = threadIdx.x;
  const int wave = tid >> 5;
  const int lane = tid & 31;
  const int b    = blockIdx.y;
  const int tile = blockIdx.x * WAVES + wave;
  const bool active = tile < ntiles;
  const int m0 = tile * 256;

  if (tid < 64) {
    const int q = tid - 16;
    float H = 0.f, G = 0.f;
    if (q >= 0 && q < 32) {
      H = ho[31 - q];
      const float hq = ho[q];
      G = (q & 1) ? -hq : hq;
    }
    sFH[tid] = H;
    sFG[tid] = G;
  }

  if (active) {
    const float* ap = apB + (size_t)b * apStride + apOff;
    const float* dg = dsB + (size_t)b * dsStride + dsOff;
    __builtin_prefetch(ap + (m0 >> 1), 0, 3);
    __builtin_prefetch(dg + (m0 >> 1), 0, 3);
    for (int s = lane; s < 288; s += 32) {
      int g = m0 + s - 16;                 // circular pad over length 2n
      if (g < 0)      g += two_n;
      if (g >= two_n) g -= two_n;
      float va = 0.f, vd = 0.f;
      if (!(g & 1)) { va = ap[g >> 1]; vd = dg[g >> 1]; }  // upsample2: odd phase = 0
      sA[wave][s] = va;
      sD[wave][s] = vd;
    }
  }
  __syncthreads();
  if (!active) return;

  const float* Sa = sA[wave];
  const float* Sd = sD[wave];
  const int row  = lane & 15;
  const int col  = row;                // shift p in [0,15]
  const int hi   = lane >> 4;
  const int koff = hi << 1;
  const int aIx  = 16 * row + koff;
  const int bIx  = 16 + koff - col;    // padded-table base (>= 1)

  v8f acc = {};
#if __has_builtin(__builtin_amdgcn_wmma_f32_16x16x4_f32)
  #pragma unroll
  for (int c = 0; c < 12; ++c) {       // Ho over approx stream
    v2f av = { Sa[aIx + 4 * c], Sa[aIx + 4 * c + 1] };
    v2f bv = { sFH[bIx + 4 * c], sFH[bIx + 4 * c + 1] };
    acc = __builtin_amdgcn_wmma_f32_16x16x4_f32(false, av, false, bv,
                                                (short)0, acc, false, false);
  }
  #pragma unroll
  for (int c = 0; c < 12; ++c) {       // Go over detail stream
    v2f av = { Sd[aIx + 4 * c], Sd[aIx + 4 * c + 1] };
    v2f bv = { sFG[bIx + 4 * c], sFG[bIx + 4 * c + 1] };
    acc = __builtin_amdgcn_wmma_f32_16x16x4_f32(false, av, false, bv,
                                                (short)0, acc, false, false);
  }
#else
  #pragma unroll
  for (int r = 0; r < 8; ++r) {
    const int M = r + 8 * hi;
    const int base = 16 * M + col;
    float sum = 0.f;
    for (int k = 0; k < 32; ++k)
      sum += Sa[base + k] * sFH[16 + k] + Sd[base + k] * sFG[16 + k];
    acc[r] = sum;
  }
#endif

  float* outP = outB + (size_t)b * outStride + outOff;
  #pragma unroll
  for (int r = 0; r < 8; ++r) {
    const int m = m0 + 16 * (r + 8 * hi) + col;
    outP[m] = acc[r];
  }
}

// ---------------------------------------------------------------------------
// Host orchestration: 8 forward + 8 inverse level launches.
//   d_out = [ energy (B*L) | appr (B*L) ]  (fp32)
//   Forward approx pyramid a1..a7 lives in the appr half (overwritten at end);
//   a8 + all details land directly in the energy half.
//   Inverse ping-pongs: even iters -> d_ws (32768 f32/batch), odd -> appr[0..],
//   final iter reads d_ws and writes the full appr region. 192MB L2 keeps the
//   whole 65MB pyramid cache-resident; HBM sees only the compulsory ~201MB.
// ---------------------------------------------------------------------------
extern "C" void kernel_launch(void* const* d_in, const int* in_sizes, int n_in,
                              void* d_out, int out_size, void* d_ws, size_t ws_size,
                              hipStream_t stream) {
  (void)in_sizes; (void)n_in; (void)out_size; (void)ws_size;
  const int Bn = 256, L = 65536;
  const float* x  = (const float*)d_in[0];
  const float* ho = (const float*)d_in[1];
  float* energy = (float*)d_out;
  float* appr   = energy + (size_t)Bn * L;
  float* ws     = (float*)d_ws;

  // -------- forward: 8 analysis levels --------
  for (int lev = 1; lev <= 8; ++lev) {
    const int Nin  = L >> (lev - 1);
    const int Nout = L >> lev;
    const float* inB; int inOff;
    if (lev == 1) { inB = x;    inOff = 0; }
    else          { inB = appr; inOff = L - (L >> (lev - 2)); }  // slot of a_{lev-1}
    const int off = L - (L >> (lev - 1));      // slot of a_lev / d_lev
    float* aB; int aOff;
    if (lev == 8) { aB = energy; aOff = off + Nout; }  // a8 -> energy tail (65280)
    else          { aB = appr;   aOff = off; }
    const int ntiles = Nout >> 7;              // 128 outputs per wave-tile
    dim3 grid((ntiles + WAVES - 1) / WAVES, Bn);
    dwt_fwd_level<<<grid, BLOCK, 0, stream>>>(inB, L, inOff,
                                              aB, L, aOff,
                                              energy, L, off,
                                              ho, Nin, ntiles);
  }

  // -------- inverse: 8 synthesis levels --------
  const float* apB = energy;            // seed = a8
  int apOff = (L - (L >> 7)) + 256;     // 65280
  int apStride = L;
  for (int i = 0; i < 8; ++i) {
    const int n = 256 << i, two_n = n << 1;
    const int dOff = L - (L >> (7 - i));        // detail level (8-i), length n
    float* oB; int oOff, oStride;
    if (i == 7)            { oB = appr; oOff = 0; oStride = L; }      // final output
    else if ((i & 1) == 0) { oB = ws;   oOff = 0; oStride = 32768; }  // even -> scratch
    else                   { oB = appr; oOff = 0; oStride = L; }      // odd  -> appr head
    const int ntiles = two_n >> 8;              // 256 outputs per wave-tile
    dim3 grid((ntiles + WAVES - 1) / WAVES, Bn);
    idwt_level<<<grid, BLOCK, 0, stream>>>(apB, apStride, apOff,
                                           energy, L, dOff,
                                           oB, oStride, oOff,
                                           ho, two_n, ntiles);
    apB = oB; apOff = oOff; apStride = oStride;
  }
}